// Convolution_12171937317098
// MI455X (gfx1250) — compile-verified
//
#include <hip/hip_runtime.h>

#define N_NODES 50000
#define N_EDGES 800000

typedef float v2f __attribute__((ext_vector_type(2)));
typedef float v8f __attribute__((ext_vector_type(8)));

// ---- CDNA5 f32 WMMA: D(16x16,f32) = A(16x4,f32) * B(4x16,f32) + C ----
// A layout (ISA 7.12.2): lanes 0-15 hold rows M=0..15 with VGPR0/1 = K0/K1,
// lanes 16-31 hold the same rows with K2/K3.  B/C/D are row-striped across
// lanes within a VGPR (VGPR v: row v for lanes 0-15, row v+half for 16-31).
__device__ __forceinline__ v8f wmma4(v2f a, v2f b, v8f c) {
  return __builtin_amdgcn_wmma_f32_16x16x4_f32(
      /*neg_a=*/false, a, /*neg_b=*/false, b,
      /*c_mod=*/(short)0, c, /*reuse_a=*/false, /*reuse_b=*/false);
}
__device__ __forceinline__ v8f zero8() {
  v8f z = {0.f, 0.f, 0.f, 0.f, 0.f, 0.f, 0.f, 0.f};
  return z;
}
// Load a 4x16 B tile slice for this lane from a row-major weight W[ld cols].
// VGPR v holds K = k4 + v (lanes 0-15) / K = k4 + v + 2 (lanes 16-31).
__device__ __forceinline__ v2f load_b(const float* __restrict__ W, int ld,
                                      int k4, int col, int khalf) {
  int k = k4 + 2 * khalf;
  v2f b;
  b.x = W[k * ld + col];
  b.y = W[(k + 1) * ld + col];
  return b;
}
__device__ __forceinline__ float gelu_t(float x) {  // jax.nn.gelu (tanh approx)
  float x3 = x * x * x;
  return 0.5f * x * (1.f + tanhf(0.7978845608028654f * (x + 0.044715f * x3)));
}

// ---------------- kernel 0: clear accumulators ----------------
__global__ void zero_kernel(float4* __restrict__ p, long n4) {
  long i = (long)blockIdx.x * blockDim.x + threadIdx.x;
  long st = (long)gridDim.x * blockDim.x;
  float4 z;
  z.x = z.y = z.z = z.w = 0.f;
  for (; i < n4; i += st) p[i] = z;
}

// ---------------- kernel 1: node-side feature linears ----------------
// f0 = node_s@W_feat0/sqrt(32) ; f1 = einsum(node_v,W_feat1)/sqrt(32)
// out <- cos(MIX) * self terms (initializes every d_out element)
__global__ __launch_bounds__(256) void node_pre_kernel(
    const float* __restrict__ node_s, const float* __restrict__ node_v,
    const float* __restrict__ Wf0, const float* __restrict__ Ws0,
    const float* __restrict__ Wf1, const float* __restrict__ Ws1,
    float* __restrict__ f0, float* __restrict__ f1, float* __restrict__ out) {
  const int lane = threadIdx.x & 31;
  const int wid = threadIdx.x >> 5;
  const int tile = blockIdx.x * 8 + wid;  // 16 nodes per wave
  if (tile * 16 >= N_NODES) return;
  const int row = lane & 15, khalf = lane >> 4;
  const int node0 = tile * 16;
  const int myrow = node0 + row;
  const float FEAT_SC = 0.17677669529663688f;                          // 1/sqrt(32)
  const float SELF_SC = 0.17677669529663688f * 0.92387953251128674f;   // *cos(pi/8)

  v2f as[8];
#pragma unroll
  for (int kc = 0; kc < 8; kc++) {
    int k = kc * 4 + 2 * khalf;
    as[kc].x = node_s[myrow * 32 + k];
    as[kc].y = node_s[myrow * 32 + k + 1];
  }
#pragma unroll
  for (int ct = 0; ct < 2; ct++) {
    v8f cf = zero8(), cs = zero8();
    int col = ct * 16 + row;
#pragma unroll
    for (int kc = 0; kc < 8; kc++) {
      cf = wmma4(as[kc], load_b(Wf0, 32, kc * 4, col, khalf), cf);
      cs = wmma4(as[kc], load_b(Ws0, 32, kc * 4, col, khalf), cs);
    }
#pragma unroll
    for (int v = 0; v < 8; v++) {
      int r = node0 + v + 8 * khalf;
      f0[r * 32 + col] = cf[v] * FEAT_SC;
      out[r * 128 + col] = cs[v] * SELF_SC;
    }
  }
#pragma unroll
  for (int comp = 0; comp < 3; comp++) {
    v2f av[8];
#pragma unroll
    for (int kc = 0; kc < 8; kc++) {
      int k = kc * 4 + 2 * khalf;
      av[kc].x = node_v[(myrow * 32 + k) * 3 + comp];
      av[kc].y = node_v[(myrow * 32 + k + 1) * 3 + comp];
    }
#pragma unroll
    for (int ct = 0; ct < 2; ct++) {
      v8f cf = zero8(), cs = zero8();
      int col = ct * 16 + row;
#pragma unroll
      for (int kc = 0; kc < 8; kc++) {
        cf = wmma4(av[kc], load_b(Wf1, 32, kc * 4, col, khalf), cf);
        cs = wmma4(av[kc], load_b(Ws1, 32, kc * 4, col, khalf), cs);
      }
#pragma unroll
      for (int v = 0; v < 8; v++) {
        int r = node0 + v + 8 * khalf;
        f1[(r * 32 + col) * 3 + comp] = cf[v] * FEAT_SC;
        out[r * 128 + 32 + col * 3 + comp] = cs[v] * SELF_SC;
      }
    }
  }
}

// ---------------- kernel 2: fused edge MLP + tensor product + scatter ----
// One wave handles 16 edges; 8 waves / block. LDS stages gelu(h) for the
// D-layout -> A-layout relayout between WMMA GEMM stages.
__global__ __launch_bounds__(256) void edge_kernel(
    const float* __restrict__ sh0, const float* __restrict__ sh1,
    const float* __restrict__ edge_scalar, const int* __restrict__ esrc,
    const int* __restrict__ edst, const float* __restrict__ mw0,
    const float* __restrict__ mw1, const float* __restrict__ wp0,
    const float* __restrict__ wp1, const float* __restrict__ wp2,
    const float* __restrict__ wp3, const float* __restrict__ f0,
    const float* __restrict__ f1, float* __restrict__ n0,
    float* __restrict__ n1) {
  __shared__ float hsh[8 * 16 * 64];  // 32 KB of the 320 KB WGP LDS
  const int lane = threadIdx.x & 31;
  const int wid = threadIdx.x >> 5;
  const int row = lane & 15, khalf = lane >> 4;
  float* hbuf = &hsh[wid * 16 * 64];
  const int ebase = (blockIdx.x * 8 + wid) * 16;
  const int erow = ebase + row;

  const float INV_RAD = 0.35355339059327373f;  // 1/sqrt(8)
  const float INV_FC = 0.125f;                 // 1/sqrt(64)
  const float INV_NN = 0.25f;                  // 1/sqrt(16)
  const float INV_S3 = 0.5773502691896258f;

  // ---- MLP layer 1: (16x8)@(8x64), gelu -> LDS ----
  v2f a1[2];
#pragma unroll
  for (int kc = 0; kc < 2; kc++) {
    int k = kc * 4 + 2 * khalf;
    a1[kc].x = edge_scalar[erow * 8 + k];
    a1[kc].y = edge_scalar[erow * 8 + k + 1];
  }
#pragma unroll
  for (int ct = 0; ct < 4; ct++) {
    v8f acc = zero8();
    int col = ct * 16 + row;
#pragma unroll
    for (int kc = 0; kc < 2; kc++)
      acc = wmma4(a1[kc], load_b(mw0, 64, kc * 4, col, khalf), acc);
#pragma unroll
    for (int v = 0; v < 8; v++)
      hbuf[(v + 8 * khalf) * 64 + col] = gelu_t(acc[v] * INV_RAD);
  }
  __syncthreads();

  // ---- MLP layer 2: (16x64)@(64x64), gelu -> LDS ----
  v8f acc2[4] = {zero8(), zero8(), zero8(), zero8()};
#pragma unroll
  for (int kc = 0; kc < 16; kc++) {
    int k = kc * 4 + 2 * khalf;
    v2f a;
    a.x = hbuf[row * 64 + k];
    a.y = hbuf[row * 64 + k + 1];
#pragma unroll
    for (int ct = 0; ct < 4; ct++)
      acc2[ct] = wmma4(a, load_b(mw1, 64, kc * 4, ct * 16 + row, khalf), acc2[ct]);
  }
  __syncthreads();
#pragma unroll
  for (int ct = 0; ct < 4; ct++)
#pragma unroll
    for (int v = 0; v < 8; v++)
      hbuf[(v + 8 * khalf) * 64 + ct * 16 + row] = gelu_t(acc2[ct][v] * INV_FC);
  __syncthreads();

  // Per-lane metadata for the 8 edge rows this lane owns in D layout.
  int srcs[8], dsts[8];
  float s0a[8], s1a[8][3];
#pragma unroll
  for (int v = 0; v < 8; v++) {
    int e = ebase + v + 8 * khalf;
    srcs[v] = esrc[e];
    dsts[v] = edst[e];
    s0a[v] = sh0[e];
    s1a[v][0] = sh1[e * 3 + 0];
    s1a[v][1] = sh1[e * 3 + 1];
    s1a[v][2] = sh1[e * 3 + 2];
  }

  // ---- projections (4x 64->32) + tensor product + scatter ----
#pragma unroll
  for (int ct = 0; ct < 2; ct++) {
    int mul = ct * 16 + row;
    v8f w0 = zero8(), w1 = zero8(), w2 = zero8(), w3 = zero8();
#pragma unroll
    for (int kc = 0; kc < 16; kc++) {
      int k = kc * 4 + 2 * khalf;
      v2f a;
      a.x = hbuf[row * 64 + k];
      a.y = hbuf[row * 64 + k + 1];
      w0 = wmma4(a, load_b(wp0, 32, kc * 4, mul, khalf), w0);
      w1 = wmma4(a, load_b(wp1, 32, kc * 4, mul, khalf), w1);
      w2 = wmma4(a, load_b(wp2, 32, kc * 4, mul, khalf), w2);
      w3 = wmma4(a, load_b(wp3, 32, kc * 4, mul, khalf), w3);
    }
#pragma unroll
    for (int v = 0; v < 8; v++) {
      int src = srcs[v], dst = dsts[v];
      float e0 = f0[src * 32 + mul];
      float e1x = f1[(src * 32 + mul) * 3 + 0];
      float e1y = f1[(src * 32 + mul) * 3 + 1];
      float e1z = f1[(src * 32 + mul) * 3 + 2];
      float w0v = w0[v] * INV_FC, w1v = w1[v] * INV_FC;
      float w2v = w2[v] * INV_FC, w3v = w3[v] * INV_FC;
      float s0s = s0a[v];
      float p0 = w0v * e0 * s0s * INV_NN;
      float dt = e1x * s1a[v][0] + e1y * s1a[v][1] + e1z * s1a[v][2];
      float p3 = w3v * dt * INV_S3 * INV_NN;
      unsafeAtomicAdd(&n0[dst * 64 + mul], p0);
      unsafeAtomicAdd(&n0[dst * 64 + 32 + mul], p3);
      float w1e0 = w1v * e0 * INV_NN;
      unsafeAtomicAdd(&n1[(dst * 64 + mul) * 3 + 0], w1e0 * s1a[v][0]);
      unsafeAtomicAdd(&n1[(dst * 64 + mul) * 3 + 1], w1e0 * s1a[v][1]);
      unsafeAtomicAdd(&n1[(dst * 64 + mul) * 3 + 2], w1e0 * s1a[v][2]);
      float w2s = w2v * s0s * INV_NN;
      unsafeAtomicAdd(&n1[(dst * 64 + 32 + mul) * 3 + 0], w2s * e1x);
      unsafeAtomicAdd(&n1[(dst * 64 + 32 + mul) * 3 + 1], w2s * e1y);
      unsafeAtomicAdd(&n1[(dst * 64 + 32 + mul) * 3 + 2], w2s * e1z);
    }
  }
}

// ---------------- kernel 3: output linears + mix ----------------
__global__ __launch_bounds__(256) void node_out_kernel(
    const float* __restrict__ n0, const float* __restrict__ n1,
    const float* __restrict__ Wo0, const float* __restrict__ Wo1,
    float* __restrict__ out) {
  const int lane = threadIdx.x & 31;
  const int wid = threadIdx.x >> 5;
  const int tile = blockIdx.x * 8 + wid;
  if (tile * 16 >= N_NODES) return;
  const int row = lane & 15, khalf = lane >> 4;
  const int node0 = tile * 16, myrow = node0 + row;
  const float SC = 0.38268343236508984f * 0.125f;  // sin(pi/8)/sqrt(64)

  v2f a0[16];
#pragma unroll
  for (int kc = 0; kc < 16; kc++) {
    int k = kc * 4 + 2 * khalf;
    a0[kc].x = n0[myrow * 64 + k];
    a0[kc].y = n0[myrow * 64 + k + 1];
  }
#pragma unroll
  for (int ct = 0; ct < 2; ct++) {
    v8f acc = zero8();
    int col = ct * 16 + row;
#pragma unroll
    for (int kc = 0; kc < 16; kc++)
      acc = wmma4(a0[kc], load_b(Wo0, 32, kc * 4, col, khalf), acc);
#pragma unroll
    for (int v = 0; v < 8; v++) {
      int r = node0 + v + 8 * khalf;
      out[r * 128 + col] += acc[v] * SC;
    }
  }
#pragma unroll
  for (int comp = 0; comp < 3; comp++) {
    v2f av[16];
#pragma unroll
    for (int kc = 0; kc < 16; kc++) {
      int k = kc * 4 + 2 * khalf;
      av[kc].x = n1[(myrow * 64 + k) * 3 + comp];
      av[kc].y = n1[(myrow * 64 + k + 1) * 3 + comp];
    }
#pragma unroll
    for (int ct = 0; ct < 2; ct++) {
      v8f acc = zero8();
      int col = ct * 16 + row;
#pragma unroll
      for (int kc = 0; kc < 16; kc++)
        acc = wmma4(av[kc], load_b(Wo1, 32, kc * 4, col, khalf), acc);
#pragma unroll
      for (int v = 0; v < 8; v++) {
        int r = node0 + v + 8 * khalf;
        out[r * 128 + 32 + col * 3 + comp] += acc[v] * SC;
      }
    }
  }
}

extern "C" void kernel_launch(void* const* d_in, const int* in_sizes, int n_in,
                              void* d_out, int out_size, void* d_ws,
                              size_t ws_size, hipStream_t stream) {
  (void)in_sizes; (void)n_in; (void)out_size; (void)ws_size;
  const float* node_s = (const float*)d_in[0];
  const float* node_v = (const float*)d_in[1];
  const float* sh0 = (const float*)d_in[2];
  const float* sh1 = (const float*)d_in[3];
  const float* edge_scalar = (const float*)d_in[4];
  const int* esrc = (const int*)d_in[5];
  const int* edst = (const int*)d_in[6];
  const float* Wf0 = (const float*)d_in[7];
  const float* Ws0 = (const float*)d_in[8];
  const float* Wf1 = (const float*)d_in[9];
  const float* Ws1 = (const float*)d_in[10];
  const float* mw0 = (const float*)d_in[11];
  const float* mw1 = (const float*)d_in[12];
  const float* wp0 = (const float*)d_in[13];
  const float* wp1 = (const float*)d_in[14];
  const float* wp2 = (const float*)d_in[15];
  const float* wp3 = (const float*)d_in[16];
  const float* Wo0 = (const float*)d_in[17];
  const float* Wo1 = (const float*)d_in[18];
  float* out = (float*)d_out;

  // workspace layout: f0 (N*32) | f1 (N*96) | n0 (N*64) | n1 (N*192) floats
  float* f0 = (float*)d_ws;
  float* f1 = f0 + (size_t)N_NODES * 32;
  float* n0 = f1 + (size_t)N_NODES * 96;
  float* n1 = n0 + (size_t)N_NODES * 64;

  long n4 = (long)N_NODES * 256 / 4;  // n0+n1 region, float4 granules
  zero_kernel<<<2048, 256, 0, stream>>>((float4*)n0, n4);

  int tiles = (N_NODES + 15) / 16;          // 3125
  int blocksA = (tiles + 7) / 8;            // 8 waves (tiles) per block
  node_pre_kernel<<<blocksA, 256, 0, stream>>>(node_s, node_v, Wf0, Ws0, Wf1,
                                               Ws1, f0, f1, out);
  edge_kernel<<<N_EDGES / 128, 256, 0, stream>>>(sh0, sh1, edge_scalar, esrc,
                                                 edst, mw0, mw1, wp0, wp1, wp2,
                                                 wp3, f0, f1, n0, n1);
  node_out_kernel<<<blocksA, 256, 0, stream>>>(n0, n1, Wo0, Wo1, out);
}